// RVQECell_90950227460858
// MI455X (gfx1250) — compile-verified
//
#include <hip/hip_runtime.h>
#include <hip/hip_bf16.h>

// RVQE cell, CDNA5 (gfx1250, wave32).
// One workgroup (128 threads, 4 waves) per batch element; state (4096 f32)
// register-resident: 32 amps/thread. Global bit b of the amplitude index:
//   bits 11..5 -> thread index t (lane l in 0..6 <-> thread bit 6-l)
//   bits 4..0  -> local index j  (lane l in 7..11 <-> local bit 11-l)
// LDS only for cross-wave gates (lanes 0,1) and for the WMMA probs tile.

typedef __attribute__((ext_vector_type(2))) float v2f;
typedef __attribute__((ext_vector_type(8))) float v8f;

#define DEVINL __device__ __forceinline__

// ---- fused rot_block: rY(sign*Phi_g) on lane 10 (local bit 1) ------------
template<int SGN>
DEVINL void rot_block(float v[32], const float cg[8], const float sg[8]) {
#pragma unroll
  for (int g = 0; g < 8; ++g) {
    float c = cg[g];
    float s = (SGN > 0) ? sg[g] : -sg[g];
#pragma unroll
    for (int b0 = 0; b0 < 2; ++b0) {
      int i0 = g * 4 + b0, i1 = i0 + 2;       // (lane10 bit 0) <-> (bit 1)
      float a = v[i0], b = v[i1];
      v[i0] = fmaf(c, a, -s * b);
      v[i1] = fmaf(s, a,  c * b);
    }
  }
}

// ---- crY(SGN*pi, control=lane10, target=lane11): signed local swap -------
template<int SGN>
DEVINL void cry_pi_10_11(float v[32]) {
#pragma unroll
  for (int g = 0; g < 8; ++g) {
    int j0 = g * 4 + 2, j1 = g * 4 + 3;       // bit1(=lane10 ctrl)==1 half
    float a = v[j0], b = v[j1];
    if (SGN > 0) { v[j0] = -b; v[j1] =  a; }
    else         { v[j0] =  b; v[j1] = -a; }
  }
}

// ---- crY(+pi, control=lane11(local bit0), target local bit M) ------------
template<int M>
DEVINL void cry_pi_11_local(float v[32]) {
#pragma unroll
  for (int j = 0; j < 32; ++j) {
    if ((j & 1) && !(j & M)) {
      float a = v[j], b = v[j | M];
      v[j] = -b; v[j | M] = a;
    }
  }
}

// ---- crY(+pi, control=lane11, target=thread bit TBIT<=4): in-wave --------
template<int TBIT>
DEVINL void cry_pi_11_wave(float v[32], int t) {
  const bool hi = (t >> TBIT) & 1;
#pragma unroll
  for (int j = 1; j < 32; j += 2) {
    float r = __shfl_xor(v[j], 1 << TBIT, 32);
    v[j] = hi ? r : -r;
  }
}

// ---- crY(+pi, control=lane11, target=thread bit TBIT in {5,6}): LDS ------
template<int TBIT>
DEVINL void cry_pi_11_x(float v[32], int t, float* smem) {
  __syncthreads();
#pragma unroll
  for (int j = 1; j < 32; j += 2) smem[(j >> 1) * 128 + t] = v[j];
  __syncthreads();
  const bool hi = (t >> TBIT) & 1;
  const int p = t ^ (1 << TBIT);
#pragma unroll
  for (int j = 1; j < 32; j += 2) {
    float r = smem[(j >> 1) * 128 + p];
    v[j] = hi ? r : -r;
  }
}

// ---- rY(theta) on a local bit (lanes 7..9) -------------------------------
template<int M>
DEVINL void ry_local(float v[32], float c, float s) {
#pragma unroll
  for (int j = 0; j < 32; ++j) {
    if (!(j & M)) {
      float a = v[j], b = v[j | M];
      v[j]     = fmaf(c, a, -s * b);
      v[j | M] = fmaf(s, a,  c * b);
    }
  }
}

// ---- rY(theta) on a thread bit, in-wave (TBIT<=4) ------------------------
template<int TBIT>
DEVINL void ry_wave(float v[32], int t, float c, float s) {
  const bool hi = (t >> TBIT) & 1;
  const float sl = hi ? s : -s;
#pragma unroll
  for (int j = 0; j < 32; ++j) {
    float r = __shfl_xor(v[j], 1 << TBIT, 32);
    v[j] = fmaf(sl, r, c * v[j]);
  }
}

// ---- rY(theta) on a thread bit, cross-wave via LDS (TBIT in {5,6}) -------
template<int TBIT>
DEVINL void ry_x(float v[32], int t, float c, float s, float* smem) {
  __syncthreads();
#pragma unroll
  for (int j = 0; j < 32; ++j) smem[j * 128 + t] = v[j];
  __syncthreads();
  const bool hi = (t >> TBIT) & 1;
  const float sl = hi ? s : -s;
  const int p = t ^ (1 << TBIT);
#pragma unroll
  for (int j = 0; j < 32; ++j)
    v[j] = fmaf(sl, smem[j * 128 + p], c * v[j]);
}

// ---- one quantum neuron: RB+ C1+ RB- C2 RB+ C1- RB- ----------------------
template<int OUT>
DEVINL void neuron(float v[32], int t, const float* th, float* smem) {
  // fused rot_block angle: base over thread-bit controls (lanes 0..6) + bias
  float base = th[10];
#pragma unroll
  for (int l = 0; l < 7; ++l)
    base += ((t >> (6 - l)) & 1) ? th[l] : 0.0f;
  const float t7 = th[7], t8 = th[8], t9 = th[9];  // local-bit controls
  float ang[8];
  ang[0] = base;        ang[1] = base + t9;
  ang[2] = base + t8;   ang[3] = ang[2] + t9;
  ang[4] = base + t7;   ang[5] = ang[4] + t9;
  ang[6] = ang[4] + t8; ang[7] = ang[6] + t9;
  float cg[8], sg[8];
#pragma unroll
  for (int g = 0; g < 8; ++g) __sincosf(0.5f * ang[g], &sg[g], &cg[g]);

  rot_block<+1>(v, cg, sg);
  cry_pi_10_11<+1>(v);
  rot_block<-1>(v, cg, sg);
  if constexpr (OUT >= 7)      cry_pi_11_local<(1 << (11 - OUT))>(v);
  else if constexpr (OUT >= 2) cry_pi_11_wave<(6 - OUT)>(v, t);
  else                         cry_pi_11_x<(6 - OUT)>(v, t, smem);
  rot_block<+1>(v, cg, sg);
  cry_pi_10_11<-1>(v);
  rot_block<-1>(v, cg, sg);
}

__global__ __launch_bounds__(128) void rvqe_kernel(
    const float* __restrict__ psi_in, const int* __restrict__ inp,
    const float* __restrict__ uth, const float* __restrict__ nth,
    float* __restrict__ out) {
  __shared__ float smem[64 * 65];   // exchange buffer (4096f) / WMMA sq tile (64x65)
  const int b = blockIdx.x;
  const int t = threadIdx.x;

  // BitFlipLayer folded into the load: flipping lane l (0..5) XORs thread bit 6-l
  int xorT = 0;
#pragma unroll
  for (int l = 0; l < 6; ++l) xorT |= (inp[l] != 0) << (6 - l);

  float v[32];
  const float* src = psi_in + ((size_t)b << 12) + (((size_t)(t ^ xorT)) << 5);
#pragma unroll
  for (int q = 0; q < 8; ++q) {
    float4 f = reinterpret_cast<const float4*>(src)[q];
    v[4 * q + 0] = f.x; v[4 * q + 1] = f.y; v[4 * q + 2] = f.z; v[4 * q + 3] = f.w;
  }

  for (int s = 0; s < 2; ++s) {
    const float* u = uth + s * 10;
    float c, sn;
    __sincosf(0.5f * u[0], &sn, &c); ry_x<6>(v, t, c, sn, smem);   // lane 0
    __sincosf(0.5f * u[1], &sn, &c); ry_x<5>(v, t, c, sn, smem);   // lane 1
    __sincosf(0.5f * u[2], &sn, &c); ry_wave<4>(v, t, c, sn);      // lane 2
    __sincosf(0.5f * u[3], &sn, &c); ry_wave<3>(v, t, c, sn);
    __sincosf(0.5f * u[4], &sn, &c); ry_wave<2>(v, t, c, sn);
    __sincosf(0.5f * u[5], &sn, &c); ry_wave<1>(v, t, c, sn);
    __sincosf(0.5f * u[6], &sn, &c); ry_wave<0>(v, t, c, sn);      // lane 6
    __sincosf(0.5f * u[7], &sn, &c); ry_local<16>(v, c, sn);       // lane 7
    __sincosf(0.5f * u[8], &sn, &c); ry_local<8>(v, c, sn);        // lane 8
    __sincosf(0.5f * u[9], &sn, &c); ry_local<4>(v, c, sn);        // lane 9

    const float* nt = nth + s * 110;
    neuron<0>(v, t, nt +  0, smem);
    neuron<1>(v, t, nt + 11, smem);
    neuron<2>(v, t, nt + 22, smem);
    neuron<3>(v, t, nt + 33, smem);
    neuron<4>(v, t, nt + 44, smem);
    neuron<5>(v, t, nt + 55, smem);
    neuron<6>(v, t, nt + 66, smem);
    neuron<7>(v, t, nt + 77, smem);
    neuron<8>(v, t, nt + 88, smem);
    neuron<9>(v, t, nt + 99, smem);
  }

  // psi output (after 512*64 probs floats)
  float* psi_out = out + 512 * 64 + ((size_t)b << 12) + ((size_t)t << 5);
#pragma unroll
  for (int q = 0; q < 8; ++q) {
    float4 f;
    f.x = v[4 * q + 0]; f.y = v[4 * q + 1]; f.z = v[4 * q + 2]; f.w = v[4 * q + 3];
    reinterpret_cast<float4*>(psi_out)[q] = f;
  }

  // ---- probs[b][x] = sum_y psi^2[x,y], x = bits 11..6, y = bits 5..0 -----
#if defined(__HIP_DEVICE_COMPILE__) && __has_builtin(__builtin_amdgcn_wmma_f32_16x16x4_f32)
  // stage squares as 64x64 tile (row = t>>1, col = (t&1)*32 + j), stride 65
  __syncthreads();
#pragma unroll
  for (int j = 0; j < 32; ++j)
    smem[(t >> 1) * 65 + (t & 1) * 32 + j] = v[j] * v[j];
  __syncthreads();
  // wave w row-sums its 16-row tile via D += A(16x4) * Ones(4x16), 16 chunks
  const int w = t >> 5, lane = t & 31;
  const int row = 16 * w + (lane & 15);
  const int koff = (lane & 16) ? 2 : 0;    // A frag: lanes 0-15 hold K0,K1; 16-31 hold K2,K3
  v8f acc = {0.f, 0.f, 0.f, 0.f, 0.f, 0.f, 0.f, 0.f};
  v2f bones; bones.x = 1.0f; bones.y = 1.0f;
#pragma unroll
  for (int c16 = 0; c16 < 16; ++c16) {
    v2f a;
    a.x = smem[row * 65 + 4 * c16 + koff + 0];
    a.y = smem[row * 65 + 4 * c16 + koff + 1];
    acc = __builtin_amdgcn_wmma_f32_16x16x4_f32(
        false, a, false, bones, (short)0, acc, false, false);
  }
  // D layout: VGPR i -> lanes 0-15: M=i, lanes 16-31: M=8+i (all N equal)
  float* probs = out + b * 64;
  if (lane == 0) {
#pragma unroll
    for (int i = 0; i < 8; ++i) probs[16 * w + i] = acc[i];
  } else if (lane == 16) {
#pragma unroll
    for (int i = 0; i < 8; ++i) probs[16 * w + 8 + i] = acc[i];
  }
#else
  // fallback: local + pairwise shuffle reduction
  float sum = 0.f;
#pragma unroll
  for (int j = 0; j < 32; ++j) sum += v[j] * v[j];
  sum += __shfl_xor(sum, 1, 32);
  if ((t & 1) == 0) out[b * 64 + (t >> 1)] = sum;
#endif
}

extern "C" void kernel_launch(void* const* d_in, const int* in_sizes, int n_in,
                              void* d_out, int out_size, void* d_ws, size_t ws_size,
                              hipStream_t stream) {
  (void)in_sizes; (void)n_in; (void)out_size; (void)d_ws; (void)ws_size;
  const float* psi = (const float*)d_in[0];
  const int*   inp = (const int*)d_in[1];
  const float* uth = (const float*)d_in[2];
  const float* nth = (const float*)d_in[3];
  float* out = (float*)d_out;
  rvqe_kernel<<<512, 128, 0, stream>>>(psi, inp, uth, nth, out);
}